// BodyKDV8_24979529793880
// MI455X (gfx1250) — compile-verified
//
#include <hip/hip_runtime.h>
#include <cstdint>

// ---------------- problem constants ----------------
#define N_VOX (96 * 96 * 96)                 // 884736 voxels per (batch,channel)
#define NCH 14
#define NBATCH 2
#define TILE_W 512                           // voxels per tile
#define TILES_PER_BATCH (N_VOX / TILE_W)     // 1728
#define GX 216                               // blocks per batch
#define TILES_PER_BLOCK (TILES_PER_BATCH / GX) // 8
#define THREADS 256                          // 8 waves (wave32)
#define BUF_FLOATS (NCH * TILE_W)            // 7168 floats
#define BUF_BYTES (BUF_FLOATS * 4)           // 28672 B
#define DYN_SMEM_BYTES (4 * BUF_BYTES)       // S0,S1,T0,T1 = 112 KB

typedef unsigned int v4u __attribute__((ext_vector_type(4)));
typedef int          v4i __attribute__((ext_vector_type(4)));
typedef int          v8i __attribute__((ext_vector_type(8)));

#if defined(__AMDGCN__) && defined(__gfx1250__) && \
    __has_builtin(__builtin_amdgcn_tensor_load_to_lds) && \
    __has_builtin(__builtin_amdgcn_s_wait_tensorcnt)
#define USE_TDM 1
#else
#define USE_TDM 0
#endif

#if USE_TDM
// Issue one TDM 2D tile load: 14 rows x TILE_W cols of f32, row stride N_VOX
// elements, from global tile start `gbase` into LDS byte offset `lds_byte_off`.
// D# packing per CDNA5 ISA ch.8 (group0 128b, group1 256b; groups 2/3 zero).
// This toolchain's builtin is the 6-arg form:
//   (uint32x4 g0, int32x8 g1, int32x4 g2, int32x4 g3, int32x8 pad, i32 cpol)
__device__ __forceinline__ void tdm_load_tile(uint32_t lds_byte_off, const float* gbase) {
  const uint64_t ga = (uint64_t)(uintptr_t)gbase;
  v4u g0;
  g0[0] = 1u;                                        // count=1 (valid user D#)
  g0[1] = lds_byte_off;                              // lds_addr [63:32]
  g0[2] = (uint32_t)ga;                              // global_addr[31:0]
  g0[3] = (uint32_t)((ga >> 32) & 0x1FFFFFFull)      // global_addr[56:32]
        | (2u << 30);                                // type=2 ("image")
  v8i g1;
  g1[0] = (int)(2u << 16);                           // data_size=4B (code 2)
  g1[1] = (int)(((uint32_t)N_VOX & 0xFFFFu) << 16);  // tensor_dim0[15:0] @ [31:16]
  g1[2] = (int)(((uint32_t)N_VOX >> 16)              // tensor_dim0[31:16]
        | ((uint32_t)NCH << 16));                    // tensor_dim1[15:0]
  g1[3] = (int)((uint32_t)TILE_W << 16);             // tile_dim0 @ [31:16]
  g1[4] = (int)NCH;                                  // tile_dim1=14, tile_dim2=0
  g1[5] = (int)N_VOX;                                // tensor_dim0_stride[31:0]
  g1[6] = 0;                                         // stride hi + dim1_stride lo
  g1[7] = 0;
  const v4i z4 = {0, 0, 0, 0};                       // groups 2/3 unused (2D)
  const v8i z8 = {0, 0, 0, 0, 0, 0, 0, 0};
  __builtin_amdgcn_tensor_load_to_lds(g0, g1, z4, z4, z8, 0);
}
#endif

__global__ void __launch_bounds__(THREADS)
kd_main(const float* __restrict__ S, const float* __restrict__ T,
        const int* __restrict__ gt,
        float* __restrict__ gsum, unsigned int* __restrict__ gcnt) {
  __shared__ float        binSum[16];
  __shared__ unsigned int binCnt[16];
  extern __shared__ float dyn[];                     // [S0|S1|T0|T1]

  const int tid   = (int)threadIdx.x;
  const int batch = (int)blockIdx.y;

  if (tid < 16) { binSum[tid] = 0.0f; binCnt[tid] = 0u; }
  __syncthreads();

  const float* Sb = S  + (size_t)batch * NCH * N_VOX;
  const float* Tb = T  + (size_t)batch * NCH * N_VOX;
  const int*   Lb = gt + (size_t)batch * N_VOX;

#if USE_TDM
  const uint32_t dynBase = __builtin_amdgcn_groupstaticsize();
  // Scalar (SGPR) wave-0 test: TDM ignores EXEC, so the issue region must be
  // *branched over* by waves 1..7, not merely masked.
  const bool issuer = (__builtin_amdgcn_readfirstlane((int)threadIdx.x) == 0);

  if (issuer) {                                      // prime buffer 0 (tile k=0)
    const int t0 = (int)blockIdx.x;
    tdm_load_tile(dynBase + 0u * BUF_BYTES, Sb + (size_t)t0 * TILE_W);
    tdm_load_tile(dynBase + 2u * BUF_BYTES, Tb + (size_t)t0 * TILE_W);
  }
  int buf = 0;
  for (int k = 0; k < TILES_PER_BLOCK; ++k) {
    const int tile = (int)blockIdx.x + k * GX;
    if (issuer) {
      if (k + 1 < TILES_PER_BLOCK) {                 // prefetch next tile
        const int nt = (int)blockIdx.x + (k + 1) * GX;
        tdm_load_tile(dynBase + (uint32_t)(buf ^ 1) * BUF_BYTES,
                      Sb + (size_t)nt * TILE_W);
        tdm_load_tile(dynBase + (uint32_t)(2 + (buf ^ 1)) * BUF_BYTES,
                      Tb + (size_t)nt * TILE_W);
        __builtin_amdgcn_s_wait_tensorcnt(2);        // current tile's 2 DMAs done
      } else {
        __builtin_amdgcn_s_wait_tensorcnt(0);
      }
    }
    __syncthreads();                                 // data visible to all waves

    // ---- compute on buffer `buf`: 2 voxels/thread, conflict-free b64 LDS ----
    const float2* s2 = (const float2*)(dyn + (size_t)buf * BUF_FLOATS);
    const float2* t2 = (const float2*)(dyn + (size_t)(2 + buf) * BUF_FLOATS);

    float mS0 = -3.0e38f, mS1 = -3.0e38f, mT0 = -3.0e38f, mT1 = -3.0e38f;
#pragma unroll
    for (int c = 0; c < NCH; ++c) {
      const float2 s = s2[c * (TILE_W / 2) + tid];
      const float2 t = t2[c * (TILE_W / 2) + tid];
      mS0 = fmaxf(mS0, s.x); mS1 = fmaxf(mS1, s.y);
      mT0 = fmaxf(mT0, t.x); mT1 = fmaxf(mT1, t.y);
    }
    float ZS0 = 0.f, ZS1 = 0.f, ZT0 = 0.f, ZT1 = 0.f, U0 = 0.f, U1 = 0.f;
#pragma unroll
    for (int c = 0; c < NCH; ++c) {
      const float2 s = s2[c * (TILE_W / 2) + tid];
      const float2 t = t2[c * (TILE_W / 2) + tid];
      ZS0 += __expf(s.x - mS0); ZS1 += __expf(s.y - mS1);
      const float e0 = __expf(t.x - mT0), e1 = __expf(t.y - mT1);
      ZT0 += e0; ZT1 += e1;
      U0 = fmaf(e0, t.x - s.x, U0); U1 = fmaf(e1, t.y - s.y, U1);
    }
    const float kl0 = U0 / ZT0 - (mT0 + __logf(ZT0)) + (mS0 + __logf(ZS0));
    const float kl1 = U1 / ZT1 - (mT1 + __logf(ZT1)) + (mS1 + __logf(ZS1));

    const int2 lab = ((const int2*)(Lb + (size_t)tile * TILE_W))[tid];
    atomicAdd(&binSum[lab.x & 15], kl0); atomicAdd(&binCnt[lab.x & 15], 1u);
    atomicAdd(&binSum[lab.y & 15], kl1); atomicAdd(&binCnt[lab.y & 15], 1u);

    __syncthreads();                                 // done before buf is reused
    buf ^= 1;
  }
#else
  // Fallback: direct global loads, values held in registers (no TDM builtin).
  for (int k = 0; k < TILES_PER_BLOCK; ++k) {
    const int tile = (int)blockIdx.x + k * GX;
    const float2* s2 = (const float2*)(Sb + (size_t)tile * TILE_W);
    const float2* t2 = (const float2*)(Tb + (size_t)tile * TILE_W);
    float2 sv[NCH], tv[NCH];
#pragma unroll
    for (int c = 0; c < NCH; ++c) {
      sv[c] = s2[(size_t)c * (N_VOX / 2) + tid];
      tv[c] = t2[(size_t)c * (N_VOX / 2) + tid];
    }
    float mS0 = -3.0e38f, mS1 = -3.0e38f, mT0 = -3.0e38f, mT1 = -3.0e38f;
#pragma unroll
    for (int c = 0; c < NCH; ++c) {
      mS0 = fmaxf(mS0, sv[c].x); mS1 = fmaxf(mS1, sv[c].y);
      mT0 = fmaxf(mT0, tv[c].x); mT1 = fmaxf(mT1, tv[c].y);
    }
    float ZS0 = 0.f, ZS1 = 0.f, ZT0 = 0.f, ZT1 = 0.f, U0 = 0.f, U1 = 0.f;
#pragma unroll
    for (int c = 0; c < NCH; ++c) {
      ZS0 += __expf(sv[c].x - mS0); ZS1 += __expf(sv[c].y - mS1);
      const float e0 = __expf(tv[c].x - mT0), e1 = __expf(tv[c].y - mT1);
      ZT0 += e0; ZT1 += e1;
      U0 = fmaf(e0, tv[c].x - sv[c].x, U0); U1 = fmaf(e1, tv[c].y - sv[c].y, U1);
    }
    const float kl0 = U0 / ZT0 - (mT0 + __logf(ZT0)) + (mS0 + __logf(ZS0));
    const float kl1 = U1 / ZT1 - (mT1 + __logf(ZT1)) + (mS1 + __logf(ZS1));
    const int2 lab = ((const int2*)(Lb + (size_t)tile * TILE_W))[tid];
    atomicAdd(&binSum[lab.x & 15], kl0); atomicAdd(&binCnt[lab.x & 15], 1u);
    atomicAdd(&binSum[lab.y & 15], kl1); atomicAdd(&binCnt[lab.y & 15], 1u);
    __syncthreads();
  }
#endif

  if (tid < 16) {
    atomicAdd(&gsum[batch * 16 + tid], binSum[tid]);
    atomicAdd(&gcnt[batch * 16 + tid], binCnt[tid]);
  }
}

// One wave32: lane i -> (batch = i>>4, class = i&15); classes 1..13 contribute
// sum/(14*count); shuffle-reduce; loss *= TAU^2(=1) * LOSS_WEIGHT(=1).
__global__ void kd_finalize(const float* __restrict__ gsum,
                            const unsigned int* __restrict__ gcnt,
                            float* __restrict__ out) {
  const int i = (int)threadIdx.x;                    // 32 lanes
  float term = 0.0f;
  const int c = i & 15;
  if (c >= 1 && c <= 13) {
    const unsigned int cnt = gcnt[i];
    if (cnt > 0u) term = gsum[i] / (14.0f * (float)cnt);
  }
  for (int off = 16; off > 0; off >>= 1)
    term += __shfl_down(term, (unsigned)off, 32);
  if (i == 0) out[0] = term;
}

extern "C" void kernel_launch(void* const* d_in, const int* in_sizes, int n_in,
                              void* d_out, int out_size, void* d_ws, size_t ws_size,
                              hipStream_t stream) {
  (void)in_sizes; (void)n_in; (void)out_size; (void)ws_size;
  const float* S  = (const float*)d_in[0];
  const float* T  = (const float*)d_in[1];
  const int*   gt = (const int*)d_in[2];
  float*        out  = (float*)d_out;
  float*        gsum = (float*)d_ws;                       // [2][16] f32
  unsigned int* gcnt = (unsigned int*)((char*)d_ws + 32 * sizeof(float));

  (void)hipMemsetAsync(d_ws, 0, 32 * (sizeof(float) + sizeof(unsigned int)), stream);
  kd_main<<<dim3(GX, NBATCH), THREADS, DYN_SMEM_BYTES, stream>>>(S, T, gt, gsum, gcnt);
  kd_finalize<<<1, 32, 0, stream>>>(gsum, gcnt, out);
}